// WindowInterAttention_81999515616105
// MI455X (gfx1250) — compile-verified
//
#include <hip/hip_runtime.h>
#include <math.h>

typedef __attribute__((ext_vector_type(2))) float v2f;
typedef __attribute__((ext_vector_type(8))) float v8f;

#define B_   32
#define H_   96
#define W_   96
#define C_   64
#define NW   4
#define KK   9   // 3x3

__device__ __forceinline__ float gelu_f(float v) {
  // exact erf-based GELU (matches torch nn.GELU default)
  return 0.5f * v * (1.0f + erff(v * 0.7071067811865475f));
}
__device__ __forceinline__ float sigmoid_f(float v) {
  return 1.0f / (1.0f + expf(-v));
}

// ---------------------------------------------------------------------------
// K1: window partition + adaptive avg pool 48x48 -> 3x3.
// grid = B*nW (128), block = 576 = 9 kk-slots * 64 channels.
// kern layout: [((b*4+w)*64 + c)*9 + ky*3+kx]
// ---------------------------------------------------------------------------
__global__ __launch_bounds__(576) void k_pool(const float* __restrict__ x,
                                              float* __restrict__ kern) {
  int blk = blockIdx.x;            // b*4 + w
  int b = blk >> 2, w = blk & 3;
  int wh = w >> 1, wl = w & 1;     // window row / col (H split first, then W)
  int tid = threadIdx.x;
  int kk = tid >> 6;               // 0..8
  int c  = tid & 63;
  int ky = kk / 3, kx = kk % 3;
  int h0 = wh * 48 + ky * 16;
  int w0 = wl * 48 + kx * 16;
  const float* xb = x + (size_t)b * H_ * W_ * C_;
  float s = 0.f;
  for (int i = 0; i < 16; ++i) {
    const float* row = xb + ((size_t)(h0 + i) * W_ + w0) * C_ + c;
    for (int j = 0; j < 16; ++j) s += row[(size_t)j * C_];
  }
  kern[((size_t)blk * C_ + c) * KK + kk] = s * (1.0f / 256.0f);
}

// ---------------------------------------------------------------------------
// K2a: SE block per (b, window): conv1x1 -> gelu -> conv1x1 -> sigmoid.
// grid = 128, block = 576.
// ---------------------------------------------------------------------------
__global__ __launch_bounds__(576) void k_se(const float* __restrict__ kern,
                                            const float* __restrict__ w1,
                                            const float* __restrict__ b1,
                                            const float* __restrict__ w2,
                                            const float* __restrict__ b2,
                                            float* __restrict__ kse) {
  __shared__ float kin[C_ * KK];
  __shared__ float t1 [C_ * KK];
  int blk = blockIdx.x;
  int tid = threadIdx.x;
  int kk = tid >> 6;       // 0..8
  int o  = tid & 63;
  kin[tid] = kern[(size_t)blk * (C_ * KK) + tid];   // flat == c*9+kk ordering
  __syncthreads();
  float s = b1[o];
  for (int i = 0; i < 64; ++i) s += w1[o * 64 + i] * kin[i * 9 + kk];
  t1[o * 9 + kk] = gelu_f(s);
  __syncthreads();
  float s2 = b2[o];
  for (int i = 0; i < 64; ++i) s2 += w2[o * 64 + i] * t1[i * 9 + kk];
  kse[(size_t)blk * (C_ * KK) + o * 9 + kk] = sigmoid_f(s2);
}

// ---------------------------------------------------------------------------
// K2b: WinKernel_Reweight -> ww (B, nW). grid = B, block = 256 (4w x 64c).
// GAP over window == mean over 3x3 pooled kernel (equal-size blocks).
// ---------------------------------------------------------------------------
__global__ __launch_bounds__(256) void k_ww(const float* __restrict__ kern,
                                            const float* __restrict__ dc_w,
                                            const float* __restrict__ dc_b,
                                            const float* __restrict__ l1_w,
                                            const float* __restrict__ l1_b,
                                            const float* __restrict__ l2_w,
                                            const float* __restrict__ l2_b,
                                            float* __restrict__ ww_out) {
  __shared__ float red[NW][64];
  __shared__ float sw[NW];
  __shared__ float hbuf[16];
  int b = blockIdx.x;
  int tid = threadIdx.x;
  int w = tid >> 6, c = tid & 63;
  const float* kp = kern + ((size_t)(b * 4 + w) * 64 + c) * 9;
  float gap = 0.f;
  for (int t = 0; t < 9; ++t) gap += kp[t];
  gap *= (1.0f / 9.0f);
  red[w][c] = dc_w[w * 64 + c] * gap;
  __syncthreads();
  if (c == 0) {
    float s = dc_b[w];
    for (int j = 0; j < 64; ++j) s += red[w][j];
    sw[w] = s;
  }
  __syncthreads();
  if (tid < 16) {
    float s = l1_b[tid];
    for (int i = 0; i < 4; ++i) s += l1_w[tid * 4 + i] * sw[i];
    hbuf[tid] = gelu_f(s);
  }
  __syncthreads();
  if (tid < 4) {
    float s = l2_b[tid];
    for (int j = 0; j < 16; ++j) s += l2_w[tid * 16 + j] * hbuf[j];
    ww_out[b * 4 + tid] = sigmoid_f(s);
  }
}

// ---------------------------------------------------------------------------
// K2c: gkf + Fusion(gk_sum) -> gsum = 2*gkf + conv1x1(fu_po, x_sum)
// (sum of the two depthwise branches, since conv is linear in the kernel).
// grid = B, block = 576. gsum layout: [(b*64 + c)*9 + kk]
// ---------------------------------------------------------------------------
__global__ __launch_bounds__(576) void k_fusion(
    const float* __restrict__ kse_g, const float* __restrict__ ww_g,
    const float* __restrict__ gkf_w, const float* __restrict__ gkf_b,
    const float* __restrict__ proj_w, const float* __restrict__ proj_b,
    const float* __restrict__ fc1_w, const float* __restrict__ fc1_b,
    const float* __restrict__ fc2_w, const float* __restrict__ fc2_b,
    const float* __restrict__ fpo_w, const float* __restrict__ fpo_b,
    float* __restrict__ gsum) {
  __shared__ float kse[NW * C_ * KK];   // [(w*64+c)*9 + kk]  == flat-channel*9+kk
  __shared__ float ww4[NW];
  __shared__ float gkf_l[C_ * KK];      // [c*9+kk]
  __shared__ float feats[C_ * NW * KK]; // [(g*4+o)*9+kk]
  __shared__ float vbuf[C_ * NW];
  __shared__ float v2buf[C_];
  __shared__ float av[NW * C_];         // [w*64+c]
  __shared__ float xsum[C_ * KK];       // [c*9+kk]
  int b = blockIdx.x, tid = threadIdx.x;
  for (int e = tid; e < NW * C_ * KK; e += 576)
    kse[e] = kse_g[(size_t)b * (NW * C_ * KK) + e];
  if (tid < 4) ww4[tid] = ww_g[b * 4 + tid];
  __syncthreads();
  int c  = tid & 63;   // doubles as o / g depending on stage
  int kk = tid >> 6;   // 0..8
  {  // gkf[c,kk] = sum_w gkf_w[c,w] * (kse * ww) + gkf_b[c]
    float s = gkf_b[c];
    for (int w = 0; w < 4; ++w)
      s += gkf_w[c * 4 + w] * kse[(w * 64 + c) * 9 + kk] * ww4[w];
    gkf_l[c * 9 + kk] = s;
  }
  {  // grouped proj: groups of 4 consecutive flat channels (g = c here)
    for (int o = 0; o < 4; ++o) {
      float s = proj_b[c * 4 + o];
      for (int i = 0; i < 4; ++i)
        s += proj_w[(c * 4 + o) * 4 + i] * kse[(c * 4 + i) * 9 + kk];
      feats[(c * 4 + o) * 9 + kk] = gelu_f(s);
    }
  }
  __syncthreads();
  if (tid < 256) {  // GAP over 3x3
    float s = 0.f;
    for (int t = 0; t < 9; ++t) s += feats[tid * 9 + t];
    vbuf[tid] = s * (1.0f / 9.0f);
  }
  __syncthreads();
  if (tid < 64) {   // fc1 (256 -> 64) + gelu
    float s = fc1_b[tid];
    for (int j = 0; j < 256; ++j) s += fc1_w[tid * 256 + j] * vbuf[j];
    v2buf[tid] = gelu_f(s);
  }
  __syncthreads();
  if (tid < 256) {  // fc2 (64 -> 256) logits, index = w*64+c
    float s = fc2_b[tid];
    for (int o = 0; o < 64; ++o) s += fc2_w[tid * 64 + o] * v2buf[o];
    av[tid] = s;
  }
  __syncthreads();
  if (tid < 64) {   // softmax over the 4 windows per channel
    float m = av[tid];
    for (int w = 1; w < 4; ++w) m = fmaxf(m, av[w * 64 + tid]);
    float e[4], den = 0.f;
    for (int w = 0; w < 4; ++w) { e[w] = expf(av[w * 64 + tid] - m); den += e[w]; }
    for (int w = 0; w < 4; ++w) av[w * 64 + tid] = e[w] / den;
  }
  __syncthreads();
  {  // x_sum[c,kk] = sum_w kse[w,c,kk] * av[w,c]
    float s = 0.f;
    for (int w = 0; w < 4; ++w) s += kse[(w * 64 + c) * 9 + kk] * av[w * 64 + c];
    xsum[c * 9 + kk] = s;
  }
  __syncthreads();
  {  // gsum = 2*gkf + (fu_po_w @ x_sum + fu_po_b)
    float s = fpo_b[c];
    for (int i = 0; i < 64; ++i) s += fpo_w[c * 64 + i] * xsum[i * 9 + kk];
    gsum[((size_t)b * 64 + c) * 9 + kk] = 2.0f * gkf_l[c * 9 + kk] + s;
  }
}

// ---------------------------------------------------------------------------
// K3: fused depthwise 3x3 (per-sample per-channel kernel) + 1x1 projection
// via V_WMMA_F32_16X16X4_F32.  grid = B * (9216/128) = 2304, block = 256.
// Each block: 128 pixels x 64 ch of y into LDS, then 8 waves x (16px x 64out)
// WMMA GEMM against po_w. EXEC is all-ones at every wmma (no divergence).
// ---------------------------------------------------------------------------
__global__ __launch_bounds__(256) void k_main(const float* __restrict__ x,
                                              const float* __restrict__ gsum,
                                              const float* __restrict__ po_w,
                                              const float* __restrict__ po_b,
                                              float* __restrict__ out) {
  __shared__ float y_lds[128 * 66];   // padded stride 66 (8B-aligned rows)
  __shared__ float pw_lds[64 * 66];   // po_w[o][i] staged
  int tid = threadIdx.x;
  int blk = blockIdx.x;
  int b = blk / 72;
  int pbase = (blk % 72) * 128;

  for (int e = tid; e < 64 * 64; e += 256)
    pw_lds[(e >> 6) * 66 + (e & 63)] = po_w[e];

  // phase 1: depthwise conv, thread = (pixel-slot tid>>6, channel tid&63)
  int c = tid & 63;
  int prow = tid >> 6;
  float k9[9];
#pragma unroll
  for (int t = 0; t < 9; ++t) k9[t] = gsum[((size_t)b * 64 + c) * 9 + t];
  const float* xb = x + (size_t)b * (H_ * W_) * C_;
  for (int it = 0; it < 32; ++it) {
    int pl = prow + it * 4;          // 0..127
    int gp = pbase + pl;
    int h = gp / 96, w = gp - h * 96;
    float acc = 0.f;
#pragma unroll
    for (int dy = -1; dy <= 1; ++dy) {
      int hh = h + dy;
      if (hh < 0 || hh >= H_) continue;
#pragma unroll
      for (int dx = -1; dx <= 1; ++dx) {
        int wc = w + dx;
        if (wc < 0 || wc >= W_) continue;
        acc += k9[(dy + 1) * 3 + (dx + 1)] * xb[((size_t)hh * W_ + wc) * C_ + c];
      }
    }
    y_lds[pl * 66 + c] = acc;
  }
  __syncthreads();

  // phase 2: per-wave 16x64 output tile = 4 N-tiles x 16 K-steps of f32 WMMA
  int wave = tid >> 5, lane = tid & 31;
  int m = lane & 15, hi = lane >> 4;
  int wpb = wave * 16;               // wave's pixel base within the block tile
  v8f acc[4];
#pragma unroll
  for (int nt = 0; nt < 4; ++nt) acc[nt] = (v8f){0, 0, 0, 0, 0, 0, 0, 0};

#pragma unroll
  for (int kk = 0; kk < 16; ++kk) {
    int c0 = kk * 4 + hi * 2;        // lanes 0-15: K=0,1 ; lanes 16-31: K=2,3
    v2f a;
    a[0] = y_lds[(wpb + m) * 66 + c0];
    a[1] = y_lds[(wpb + m) * 66 + c0 + 1];
#pragma unroll
    for (int nt = 0; nt < 4; ++nt) {
      v2f bf;                        // B[k][n] = po_w[n][k]
      bf[0] = pw_lds[(nt * 16 + m) * 66 + c0];
      bf[1] = pw_lds[(nt * 16 + m) * 66 + c0 + 1];
      acc[nt] = __builtin_amdgcn_wmma_f32_16x16x4_f32(
          false, a, false, bf, (short)0, acc[nt], false, false);
    }
  }

  // store: D vgpr r -> row r (lanes 0-15) / r+8 (lanes 16-31), col = lane%16
  float* ob = out + ((size_t)b * (H_ * W_) + pbase + wpb) * C_;
#pragma unroll
  for (int nt = 0; nt < 4; ++nt) {
    float bias = po_b[nt * 16 + m];
#pragma unroll
    for (int r = 0; r < 8; ++r) {
      int row = r + hi * 8;
      ob[(size_t)row * C_ + nt * 16 + m] = acc[nt][r] + bias;
    }
  }
}

// ---------------------------------------------------------------------------
extern "C" void kernel_launch(void* const* d_in, const int* in_sizes, int n_in,
                              void* d_out, int out_size, void* d_ws, size_t ws_size,
                              hipStream_t stream) {
  const float* x       = (const float*)d_in[0];
  const float* se_w1   = (const float*)d_in[1];
  const float* se_b1   = (const float*)d_in[2];
  const float* se_w2   = (const float*)d_in[3];
  const float* se_b2   = (const float*)d_in[4];
  const float* dc_w    = (const float*)d_in[5];
  const float* dc_b    = (const float*)d_in[6];
  const float* l1_w    = (const float*)d_in[7];
  const float* l1_b    = (const float*)d_in[8];
  const float* l2_w    = (const float*)d_in[9];
  const float* l2_b    = (const float*)d_in[10];
  const float* gkf_w   = (const float*)d_in[11];
  const float* gkf_b   = (const float*)d_in[12];
  const float* fu_proj_w = (const float*)d_in[13];
  const float* fu_proj_b = (const float*)d_in[14];
  const float* fu_fc1_w  = (const float*)d_in[15];
  const float* fu_fc1_b  = (const float*)d_in[16];
  const float* fu_fc2_w  = (const float*)d_in[17];
  const float* fu_fc2_b  = (const float*)d_in[18];
  const float* fu_po_w   = (const float*)d_in[19];
  const float* fu_po_b   = (const float*)d_in[20];
  const float* po_w      = (const float*)d_in[21];
  const float* po_b      = (const float*)d_in[22];
  float* out = (float*)d_out;

  float* ws      = (float*)d_ws;
  float* ws_kern = ws;                 // B*nW*C*9  = 73728
  float* ws_kse  = ws + 73728;         // 73728
  float* ws_ww   = ws + 147456;        // 128
  float* ws_gsum = ws + 147584;        // B*C*9 = 18432

  k_pool<<<B_ * NW, 576, 0, stream>>>(x, ws_kern);
  k_se<<<B_ * NW, 576, 0, stream>>>(ws_kern, se_w1, se_b1, se_w2, se_b2, ws_kse);
  k_ww<<<B_, 256, 0, stream>>>(ws_kern, dc_w, dc_b, l1_w, l1_b, l2_w, l2_b, ws_ww);
  k_fusion<<<B_, 576, 0, stream>>>(ws_kse, ws_ww, gkf_w, gkf_b,
                                   fu_proj_w, fu_proj_b, fu_fc1_w, fu_fc1_b,
                                   fu_fc2_w, fu_fc2_b, fu_po_w, fu_po_b, ws_gsum);
  k_main<<<B_ * 72, 256, 0, stream>>>(x, ws_gsum, po_w, po_b, out);
}